// PyGCompatibleGCN_61864708932305
// MI455X (gfx1250) — compile-verified
//
#include <hip/hip_runtime.h>
#include <hip/hip_bf16.h>
#include <cstdint>

// ---------------------------------------------------------------------------
// GCN: h1 = relu(Agg(x@W1)+b1); out = log_softmax(Agg(h1@W2)+b2)
// Agg(h)[v] = sum_{e: dst=v} h[src_e]*dinv[src_e]*dinv[v] + 2*h[v]*dinv[v]^2
// deg[v] = #incoming edges + 2 (two concatenated self-loop sets in reference)
// ---------------------------------------------------------------------------

typedef __attribute__((ext_vector_type(2))) float v2f;
typedef __attribute__((ext_vector_type(8))) float v8f;

#define IN_CH 128

// ---------------- degree / dinv ----------------
__global__ void degree_kernel(const long long* __restrict__ dst,
                              float* __restrict__ deg, int nEdges) {
    int e = blockIdx.x * blockDim.x + threadIdx.x;
    if (e < nEdges) atomicAdd(&deg[(size_t)dst[e]], 1.0f);
}

__global__ void dinv_kernel(float* __restrict__ deg_dinv, int n) {
    int v = blockIdx.x * blockDim.x + threadIdx.x;
    if (v < n) {
        float d = deg_dinv[v] + 2.0f;   // +2 self loops; always > 0
        deg_dinv[v] = rsqrtf(d);
    }
}

// ---------------- GEMM via V_WMMA_F32_16X16X4_F32 ----------------
// C[rows x NCOLS] = X[rows x 128] @ W[128 x NCOLS], W staged in LDS.
// 8 waves/block, one 16-row tile per wave.
// Loop order: K outer, colTile inner -> NCOLS/16 live accumulators, ONE live
// A fragment (prevents the af[] scratch spills seen in round 1).
template <int NCOLS>
__global__ void gemm_wmma_f32(const float* __restrict__ X,
                              const float* __restrict__ W,
                              float* __restrict__ C, int rowTiles) {
    constexpr int CT = NCOLS / 16;          // column tiles (8 or 4)
    __shared__ float sW[IN_CH * NCOLS];
    const int tid = threadIdx.x;
    for (int i = tid; i < IN_CH * NCOLS; i += 256) sW[i] = W[i];
    __syncthreads();

    const int wave = tid >> 5;
    const int lane = tid & 31;
    const int rowTile = blockIdx.x * 8 + wave;
    if (rowTile >= rowTiles) return;

    const int mrow = rowTile * 16 + (lane & 15);   // A-frag row for this lane
    const int koff = (lane >> 4) * 2;              // lanes 0-15: K{0,1}; 16-31: K{2,3}
    const int nlane = lane & 15;
    const float* xrow = X + (size_t)mrow * IN_CH + koff;   // 8B aligned
    __builtin_prefetch(xrow, 0, 1);                // global_prefetch_b8

    v8f acc[CT];
#pragma unroll
    for (int ct = 0; ct < CT; ++ct) acc[ct] = (v8f){};

#pragma unroll 4
    for (int kk = 0; kk < IN_CH / 4; ++kk) {
        const int k = kk * 4 + koff;
        const v2f a = *(const v2f*)(xrow + kk * 4);        // one live A frag
#pragma unroll
        for (int ct = 0; ct < CT; ++ct) {
            v2f b;                                          // B 4x16 frag from LDS
            b[0] = sW[(k + 0) * NCOLS + ct * 16 + nlane];
            b[1] = sW[(k + 1) * NCOLS + ct * 16 + nlane];
            acc[ct] = __builtin_amdgcn_wmma_f32_16x16x4_f32(
                false, a, false, b, (short)0, acc[ct], false, false);
        }
    }

    // C/D layout: VGPR r -> row rowTile*16 + r + 8*(lane>=16), col = ct*16+nlane
    const int mBase = rowTile * 16 + 8 * (lane >> 4);
#pragma unroll
    for (int ct = 0; ct < CT; ++ct) {
        const int ncol = ct * 16 + nlane;
#pragma unroll
        for (int r = 0; r < 8; ++r)
            C[(size_t)(mBase + r) * NCOLS + ncol] = acc[ct][r];
    }
}

// ---------------- edge gather-scale-scatter (atomics into L2) ----------------
// One wave per edge; CPL = channels per lane (C = 32*CPL).
template <int CPL>
__global__ void edge_scatter(const long long* __restrict__ src,
                             const long long* __restrict__ dst,
                             const float* __restrict__ dinv,
                             const float* __restrict__ h,
                             float* __restrict__ out, int nEdges) {
    const int C = CPL * 32;
    const int e = blockIdx.x * (blockDim.x >> 5) + (threadIdx.x >> 5);
    if (e >= nEdges) return;
    const int lane = threadIdx.x & 31;
    const size_t s = (size_t)src[e];
    const size_t d = (size_t)dst[e];
    const float nrm = dinv[s] * dinv[d];
    const float* hp = h + s * C + lane * CPL;      // coalesced 512B/256B per wave
    float* op = out + d * C + lane * CPL;
    float hv[CPL];
#pragma unroll
    for (int i = 0; i < CPL; ++i) hv[i] = hp[i];
#pragma unroll
    for (int i = 0; i < CPL; ++i) atomicAdd(&op[i], hv[i] * nrm);
}

// ---------------- layer-1 finalize: self loops + bias + relu ----------------
__global__ void finalize_relu(const float* __restrict__ h,
                              const float* __restrict__ dinv,
                              const float* __restrict__ bias,
                              float* __restrict__ a, int nNodes) {
    const int idx = blockIdx.x * blockDim.x + threadIdx.x;
    const int v = idx >> 5;                // 32 lanes x 4 ch = 128
    if (v >= nNodes) return;
    const int lane = idx & 31;
    const float di = dinv[v];
    const float sl = 2.0f * di * di;
    const int c0 = lane * 4;
    const float4 hv = *(const float4*)(h + (size_t)v * 128 + c0);
    const float4 bv = *(const float4*)(bias + c0);
    float4* ap = (float4*)(a + (size_t)v * 128 + c0);
    float4 av = *ap;
    av.x = fmaxf(av.x + sl * hv.x + bv.x, 0.0f);
    av.y = fmaxf(av.y + sl * hv.y + bv.y, 0.0f);
    av.z = fmaxf(av.z + sl * hv.z + bv.z, 0.0f);
    av.w = fmaxf(av.w + sl * hv.w + bv.w, 0.0f);
    *ap = av;
}

// ---------------- layer-2 finalize: self loops + bias + log_softmax ----------
__global__ void finalize_logsoftmax(const float* __restrict__ h,
                                    const float* __restrict__ dinv,
                                    const float* __restrict__ bias,
                                    float* __restrict__ out, int nNodes) {
    const int v = blockIdx.x * (blockDim.x >> 5) + (threadIdx.x >> 5);
    if (v >= nNodes) return;
    const int lane = threadIdx.x & 31;     // 32 lanes x 2 ch = 64
    const float di = dinv[v];
    const float sl = 2.0f * di * di;
    const int c0 = lane * 2;
    const float2 hv = *(const float2*)(h + (size_t)v * 64 + c0);
    const float2 bv = *(const float2*)(bias + c0);
    float2* op = (float2*)(out + (size_t)v * 64 + c0);
    float2 ov = *op;
    float x0 = ov.x + sl * hv.x + bv.x;
    float x1 = ov.y + sl * hv.y + bv.y;
    // wave32 cross-lane max / sum-exp over 64 values
    float mx = fmaxf(x0, x1);
#pragma unroll
    for (int o = 16; o > 0; o >>= 1) mx = fmaxf(mx, __shfl_xor(mx, o, 32));
    float se = __expf(x0 - mx) + __expf(x1 - mx);
#pragma unroll
    for (int o = 16; o > 0; o >>= 1) se += __shfl_xor(se, o, 32);
    const float lse = mx + __logf(se);
    ov.x = x0 - lse;
    ov.y = x1 - lse;
    *op = ov;
}

// ---------------------------------------------------------------------------
extern "C" void kernel_launch(void* const* d_in, const int* in_sizes, int n_in,
                              void* d_out, int out_size, void* d_ws, size_t ws_size,
                              hipStream_t stream) {
    const float* x = (const float*)d_in[0];
    const long long* ei = (const long long*)d_in[1];   // int64 per reference
    const float* W1 = (const float*)d_in[2];
    const float* b1 = (const float*)d_in[3];
    const float* W2 = (const float*)d_in[4];
    const float* b2 = (const float*)d_in[5];
    float* out = (float*)d_out;

    const int N = in_sizes[0] / IN_CH;       // 100000
    const int E = in_sizes[1] / 2;           // 1600000
    const long long* srcA = ei;
    const long long* dstA = ei + E;

    // workspace: dinv[N] | h[N*128] | accum[N*128]
    char* ws = (char*)d_ws;
    size_t off0 = 0;
    size_t off1 = ((size_t)N * 4 + 255) & ~(size_t)255;
    size_t off2 = off1 + (((size_t)N * 128 * 4 + 255) & ~(size_t)255);
    float* dinv = (float*)(ws + off0);
    float* hbuf = (float*)(ws + off1);       // h1, later reused for h2 (N*64)
    float* abuf = (float*)(ws + off2);       // layer-1 aggregation

    hipMemsetAsync(dinv, 0, (size_t)N * 4, stream);
    hipMemsetAsync(abuf, 0, (size_t)N * 128 * 4, stream);
    hipMemsetAsync(out, 0, (size_t)out_size * 4, stream);

    // degrees -> dinv
    degree_kernel<<<(E + 255) / 256, 256, 0, stream>>>(dstA, dinv, E);
    dinv_kernel<<<(N + 255) / 256, 256, 0, stream>>>(dinv, N);

    const int rowTiles = (N + 15) / 16;      // 6250 (N divisible by 16)
    const int gemmBlocks = (rowTiles + 7) / 8;

    // layer 1
    gemm_wmma_f32<128><<<gemmBlocks, 256, 0, stream>>>(x, W1, hbuf, rowTiles);
    edge_scatter<4><<<(E + 7) / 8, 256, 0, stream>>>(srcA, dstA, dinv, hbuf, abuf, E);
    finalize_relu<<<((size_t)N * 32 + 255) / 256, 256, 0, stream>>>(hbuf, dinv, b1, abuf, N);

    // layer 2 (h2 reuses hbuf: N*64 <= N*128)
    gemm_wmma_f32<64><<<gemmBlocks, 256, 0, stream>>>(abuf, W2, hbuf, rowTiles);
    edge_scatter<2><<<(E + 7) / 8, 256, 0, stream>>>(srcA, dstA, dinv, hbuf, out, E);
    finalize_logsoftmax<<<((size_t)N * 32 + 255) / 256, 256, 0, stream>>>(hbuf, dinv, b2, out, N);
}